// Encoding_88613765251683
// MI455X (gfx1250) — compile-verified
//
#include <hip/hip_runtime.h>

#define D_   128
#define K_   32
#define B_   16
#define N_   4096   // H*W
#define TN   128    // pixels per workgroup chunk
#define BLOCK 256   // 8 wave32
#define XP   132    // LDS pitch for Xs [d][p]  (128 data + 4 pad, via TDM pad)
#define CP   36     // LDS pitch for Cs [d][c]
#define AP   40     // LDS pitch for As [p][c]

typedef __attribute__((ext_vector_type(2))) float v2f;
typedef __attribute__((ext_vector_type(8))) float v8f;
typedef __attribute__((ext_vector_type(4))) unsigned int v4u;
typedef __attribute__((ext_vector_type(4))) int v4i;
typedef __attribute__((ext_vector_type(8))) int v8i;

__global__ __launch_bounds__(256) void enc_zero(float* __restrict__ out,
                                                float* __restrict__ asum) {
  int i = blockIdx.x * 256 + threadIdx.x;
  if (i < B_ * K_ * D_) out[i] = 0.0f;
  if (i < B_ * K_)      asum[i] = 0.0f;
}

__global__ __launch_bounds__(BLOCK) void enc_main(const float* __restrict__ x,
                                                  const float* __restrict__ cw,
                                                  const float* __restrict__ sc,
                                                  float* __restrict__ out,
                                                  float* __restrict__ asum_g) {
  extern __shared__ float smem[];
  float* Xs    = smem;             // [D_][XP]  X chunk, [d][p], TDM-padded rows
  float* Cs    = Xs + D_ * XP;     // [D_][CP]  codewords, [d][c]
  float* As    = Cs + D_ * CP;     // [TN][AP]  softmax probs, [p][c]
  float* x2s   = As + TN * AP;     // [TN]
  float* c2s   = x2s + TN;         // [K_]
  float* scs   = c2s + K_;         // [K_]
  float* asums = scs + K_;         // [K_]

  const int tid = threadIdx.x;
  const int b   = blockIdx.x >> 5;          // 32 chunks per batch
  const int n0  = (blockIdx.x & 31) * TN;

  // ---- stage 0a: X chunk via Tensor Data Mover ----------------------------
  // 2D tile: dim0 = 128 pixels (contiguous, 512B rows), dim1 = 128 features,
  // row stride N_ elements; LDS padding 4 DWORDs per 128 DWORDs -> pitch 132.
  const float* xb = x + (size_t)b * D_ * N_ + n0;
  if (tid < 32) {
    unsigned long long ga = (unsigned long long)(uintptr_t)xb;
    unsigned int lds_off  = (unsigned int)(uintptr_t)Xs;   // addr[31:0] = LDS offset
    v4u g0;
    g0[0] = 1u;                                  // count=1, user descriptor
    g0[1] = lds_off;                             // lds_addr
    g0[2] = (unsigned int)ga;                    // global_addr[31:0]
    g0[3] = (unsigned int)((ga >> 32) & 0x01FFFFFFu) | 0x80000000u; // ga[56:32] | type=2
    v8i g1;
    g1[0] = (int)((2u << 16)     // data_size = 4B
                | (1u << 20)     // pad_enable
                | (6u << 22)     // pad_interval: every 128 DWORDs
                | (3u << 25));   // pad_amount: 4 DWORDs
    g1[1] = (int)((unsigned)N_ << 16);      // tensor_dim0[15:0]=4096 (<<16)
    g1[2] = (int)((unsigned)D_ << 16);      // tensor_dim0[31:16]=0 | tensor_dim1 lo=128
    g1[3] = (int)((unsigned)TN << 16);      // tensor_dim1 hi=0 | tile_dim0=128
    g1[4] = (int)(unsigned)D_;              // tile_dim1=128 | tile_dim2=0
    g1[5] = (int)(unsigned)N_;              // tensor_dim0_stride[31:0]=4096
    g1[6] = 0;                              // stride hi / dim1_stride lo
    g1[7] = 0;
    v4i gz = {0, 0, 0, 0};
#if defined(__clang_major__) && (__clang_major__ >= 23)
    v8i gz8 = {0, 0, 0, 0, 0, 0, 0, 0};
    __builtin_amdgcn_tensor_load_to_lds(g0, g1, gz, gz, gz8, 0);
#else
    __builtin_amdgcn_tensor_load_to_lds(g0, g1, gz, gz, 0);
#endif
  }

  // ---- stage 0b: codewords -> LDS (transposed to [d][c]) ------------------
  for (int i = tid; i < K_ * (D_ / 4); i += BLOCK) {
    int c  = i >> 5;                 // D/4 == 32
    int d4 = (i & 31) << 2;
    float4 v = *(const float4*)(cw + c * D_ + d4);
    Cs[(d4 + 0) * CP + c] = v.x;
    Cs[(d4 + 1) * CP + c] = v.y;
    Cs[(d4 + 2) * CP + c] = v.z;
    Cs[(d4 + 3) * CP + c] = v.w;
  }
  if (tid < K_) { scs[tid] = sc[tid]; asums[tid] = 0.0f; }
  if (tid < 32) __builtin_amdgcn_s_wait_tensorcnt(0);   // issuing wave drains TDM
  __syncthreads();

  // ---- x2 per pixel, c2 per codeword ----
  if (tid < TN) {
    float s = 0.0f;
    for (int d = 0; d < D_; ++d) { float v = Xs[d * XP + tid]; s += v * v; }
    x2s[tid] = s;
  } else if (tid < TN + K_) {
    int c = tid - TN;
    float s = 0.0f;
    for (int d = 0; d < D_; ++d) { float v = Cs[d * CP + c]; s += v * v; }
    c2s[c] = s;
  }
  __syncthreads();

  const int wave = tid >> 5;        // 8 waves; wave owns pixels [wave*16, +16)
  const int lane = tid & 31;
  const int half = lane >> 4;       // 0: K-lo rows, 1: K-hi rows of fragments
  const int l15  = lane & 15;

  // ---- phase 1: xc tile = X(16xD) * C^T(DxK) via f32 WMMA, k-step = 4 ----
  v8f acc0 = {};                    // codewords 0..15
  v8f acc1 = {};                    // codewords 16..31
  const int mrow = wave * 16 + l15; // pixel index for A-fragment
  for (int d0 = 0; d0 < D_; d0 += 4) {
    int dk = d0 + half * 2;
    v2f a, b0, b1;
    a.x  = Xs[dk * XP + mrow];
    a.y  = Xs[(dk + 1) * XP + mrow];
    b0.x = Cs[dk * CP + l15];
    b0.y = Cs[(dk + 1) * CP + l15];
    b1.x = Cs[dk * CP + 16 + l15];
    b1.y = Cs[(dk + 1) * CP + 16 + l15];
    acc0 = __builtin_amdgcn_wmma_f32_16x16x4_f32(false, a, false, b0,
                                                 (short)0, acc0, false, false);
    acc1 = __builtin_amdgcn_wmma_f32_16x16x4_f32(false, a, false, b1,
                                                 (short)0, acc1, false, false);
  }

  // ---- scaled-L2 + softmax over K (32 values spread over 16 lanes x 2 accs) --
  const float sc0 = scs[l15], sc1 = scs[l15 + 16];
  const float cc0 = c2s[l15], cc1 = c2s[l15 + 16];
  float t0 = 0.0f, t1 = 0.0f;       // per-codeword asum partials
#pragma unroll
  for (int r = 0; r < 8; ++r) {
    int p = wave * 16 + half * 8 + r;      // D-matrix row -> pixel
    float xv  = x2s[p];
    float sl0 = sc0 * (xv - 2.0f * acc0[r] + cc0);
    float sl1 = sc1 * (xv - 2.0f * acc1[r] + cc1);
    float m = fmaxf(sl0, sl1);
#pragma unroll
    for (int off = 1; off < 16; off <<= 1) m = fmaxf(m, __shfl_xor(m, off, 32));
    float e0 = __expf(sl0 - m);
    float e1 = __expf(sl1 - m);
    float s  = e0 + e1;
#pragma unroll
    for (int off = 1; off < 16; off <<= 1) s += __shfl_xor(s, off, 32);
    float inv = 1.0f / s;
    e0 *= inv; e1 *= inv;
    As[p * AP + l15]      = e0;
    As[p * AP + 16 + l15] = e1;
    t0 += e0; t1 += e1;
  }
  // fold the two 8-pixel halves, then accumulate per-codeword sums in LDS
  t0 += __shfl_xor(t0, 16, 32);
  t1 += __shfl_xor(t1, 16, 32);
  if (lane < 16) {
    atomicAdd(&asums[l15], t0);
    atomicAdd(&asums[l15 + 16], t1);
  }
  __syncthreads();

  // ---- phase 2: E_partial(KxD) += A^T(KxTN) * X(TNxD), k-step = 4 pixels ----
  for (int t = wave; t < 16; t += 8) {
    int mt = t & 1;                  // codeword tile (2 of 16)
    int nt = t >> 1;                 // feature tile (8 of 16)
    int drow = nt * 16 + l15;        // feature column for B/D fragments
    v8f acc = {};
    for (int ks = 0; ks < TN / 4; ++ks) {
      int p = ks * 4 + half * 2;
      v2f a, bf;
      a.x  = As[p * AP + mt * 16 + l15];
      a.y  = As[(p + 1) * AP + mt * 16 + l15];
      bf.x = Xs[drow * XP + p];
      bf.y = Xs[drow * XP + p + 1];
      acc = __builtin_amdgcn_wmma_f32_16x16x4_f32(false, a, false, bf,
                                                  (short)0, acc, false, false);
    }
    float* ob = out + ((size_t)b * K_ + mt * 16 + half * 8) * D_ + drow;
#pragma unroll
    for (int r = 0; r < 8; ++r) atomicAdd(ob + r * D_, acc[r]);
  }

  if (tid < K_) atomicAdd(&asum_g[b * K_ + tid], asums[tid]);
}

__global__ __launch_bounds__(256) void enc_final(float* __restrict__ out,
                                                 const float* __restrict__ asum,
                                                 const float* __restrict__ cw) {
  int i = blockIdx.x * 256 + threadIdx.x;
  if (i >= B_ * K_ * D_) return;
  int d = i & (D_ - 1);
  int k = (i >> 7) & (K_ - 1);
  out[i] -= asum[i >> 7] * cw[k * D_ + d];   // i>>7 == b*K_ + k
}

extern "C" void kernel_launch(void* const* d_in, const int* in_sizes, int n_in,
                              void* d_out, int out_size, void* d_ws, size_t ws_size,
                              hipStream_t stream) {
  (void)in_sizes; (void)n_in; (void)out_size; (void)ws_size;
  const float* x  = (const float*)d_in[0];
  const float* cw = (const float*)d_in[1];
  const float* sc = (const float*)d_in[2];
  float* out  = (float*)d_out;
  float* asum = (float*)d_ws;     // B_*K_ floats of scratch

  enc_zero<<<(B_ * K_ * D_ + 255) / 256, 256, 0, stream>>>(out, asum);

  size_t smem = (size_t)(D_ * XP + D_ * CP + TN * AP + TN + 3 * K_) * sizeof(float);
  enc_main<<<B_ * (N_ / TN), BLOCK, smem, stream>>>(x, cw, sc, out, asum);

  enc_final<<<(B_ * K_ * D_ + 255) / 256, 256, 0, stream>>>(out, asum, cw);
}